// GCLSTM_RGCN_89008902243182
// MI455X (gfx1250) — compile-verified
//
#include <hip/hip_runtime.h>
#include <hip/hip_bf16.h>

#define NN 100000
#define EE 1600000
#define FF 128
#define SS 256
#define CC 32

typedef float v2f __attribute__((ext_vector_type(2)));
typedef float v8f __attribute__((ext_vector_type(8)));

static __device__ __forceinline__ v8f wmma_f32(v2f a, v2f b, v8f c) {
  // D = A(16x4,f32) x B(4x16,f32) + C(16x16,f32)  -> v_wmma_f32_16x16x4_f32
  return __builtin_amdgcn_wmma_f32_16x16x4_f32(false, a, false, b, (short)0, c,
                                               false, false);
}

static __device__ __forceinline__ float sigf(float x) {
  return 1.0f / (1.0f + __expf(-x));
}

// ---------------- degree / normalization ----------------
__global__ void k_deg_init(float* deg) {
  int i = blockIdx.x * 256 + threadIdx.x;
  if (i < NN) deg[i] = 1.0f;  // self-loop weight
}

__global__ void k_deg_acc(const int* __restrict__ dst,
                          const float* __restrict__ w, float* deg) {
  int e = blockIdx.x * 256 + threadIdx.x;
  if (e < EE) atomicAdd(&deg[dst[e]], w[e]);
}

__global__ void k_dinv(float* deg) {
  int i = blockIdx.x * 256 + threadIdx.x;
  if (i < NN) {
    float d = deg[i];
    deg[i] = d > 0.0f ? rsqrtf(d) : 0.0f;
  }
}

// ---------------- GEMM1: xw = x @ W1  (N x 128) x (128 x 128) ----------------
__global__ __launch_bounds__(256) void k_gemm_xw(const float* __restrict__ x,
                                                 const float* __restrict__ W1,
                                                 float* __restrict__ xw) {
  __shared__ float As[16][132];
  const int tid = threadIdx.x;
  const int row0 = blockIdx.x * 16;
  for (int i = tid; i < 16 * FF; i += 256)
    As[i >> 7][i & 127] = x[(size_t)(row0 + (i >> 7)) * FF + (i & 127)];
  __syncthreads();

  const int lane = tid & 31, wv = tid >> 5;
  const int col = wv * 16 + (lane & 15);
  const int arow = lane & 15;
  const int krow = 2 * (lane >> 4);
  v8f acc = {};
  for (int kk = 0; kk < FF; kk += 4) {
    const int kb = kk + krow;
    v2f a; a.x = As[arow][kb]; a.y = As[arow][kb + 1];
    v2f b; b.x = W1[kb * FF + col]; b.y = W1[(kb + 1) * FF + col];
    acc = wmma_f32(a, b, acc);
  }
  const int rbase = 8 * (lane >> 4);
#pragma unroll
  for (int r = 0; r < 8; r++)
    xw[(size_t)(row0 + rbase + r) * FF + col] = acc[r];
}

// ---------------- h = b1 + dinv^2 * xw (self loop) ----------------
__global__ void k_h_init(const float* __restrict__ xw,
                         const float* __restrict__ dinv,
                         const float* __restrict__ b1, float* __restrict__ h) {
  int idx = blockIdx.x * 256 + threadIdx.x;
  if (idx < NN * FF) {
    int n = idx >> 7, f = idx & 127;
    float dv = dinv[n];
    h[idx] = b1[f] + dv * dv * xw[idx];
  }
}

// ---------------- edge scatter: h[dst] += xw[src] * norm ----------------
__global__ __launch_bounds__(256) void k_scatter(
    const int* __restrict__ src, const int* __restrict__ dst,
    const float* __restrict__ w, const float* __restrict__ dinv,
    const float* __restrict__ xw, float* __restrict__ h) {
  const int e = blockIdx.x * 8 + (threadIdx.x >> 5);
  const int lane = threadIdx.x & 31;
  const int s = src[e], d = dst[e];
  const float nrm = dinv[s] * w[e] * dinv[d];
  const float4 v = ((const float4*)(xw + (size_t)s * FF))[lane];
  float* hp = h + (size_t)d * FF + lane * 4;
  atomicAdd(hp + 0, v.x * nrm);
  atomicAdd(hp + 1, v.y * nrm);
  atomicAdd(hp + 2, v.z * nrm);
  atomicAdd(hp + 3, v.w * nrm);
}

// ---------------- fused gates: Hn, Cn ----------------
__global__ __launch_bounds__(256) void k_gates(
    const float* __restrict__ h, const float* __restrict__ H0,
    const float* __restrict__ C0, const float* __restrict__ Wi,
    const float* __restrict__ Wf, const float* __restrict__ Wc,
    const float* __restrict__ Wo, const float* __restrict__ Ti,
    const float* __restrict__ Tf, const float* __restrict__ Tc,
    const float* __restrict__ To, const float* __restrict__ bci,
    const float* __restrict__ bcf, const float* __restrict__ bcc,
    const float* __restrict__ bco, const float* __restrict__ bi,
    const float* __restrict__ bfg, const float* __restrict__ bcg,
    const float* __restrict__ bo, float* __restrict__ HnOut,
    float* __restrict__ CnOut) {
  __shared__ float Hs[16][132];
  __shared__ float Gs[16][260];
  const int tid = threadIdx.x;
  const int row0 = blockIdx.x * 16;
  for (int i = tid; i < 16 * FF; i += 256)
    Hs[i >> 7][i & 127] = h[(size_t)(row0 + (i >> 7)) * FF + (i & 127)];
  for (int i = tid; i < 16 * SS; i += 256)
    Gs[i >> 8][i & 255] = H0[(size_t)(row0 + (i >> 8)) * SS + (i & 255)];
  __syncthreads();

  const int lane = tid & 31, wv = tid >> 5;
  const int arow = lane & 15;
  const int krow = 2 * (lane >> 4);
  const int c0 = wv * 32 + (lane & 15);
  const int c1 = c0 + 16;
  const float* Wg[4] = {Wi, Wf, Wc, Wo};
  const float* Tg[4] = {Ti, Tf, Tc, To};
  v8f acc[4][2] = {};

  for (int kk = 0; kk < FF; kk += 4) {  // h @ W_g
    const int kb = kk + krow;
    v2f a; a.x = Hs[arow][kb]; a.y = Hs[arow][kb + 1];
#pragma unroll
    for (int g = 0; g < 4; g++) {
      const float* W = Wg[g];
      v2f b0; b0.x = W[kb * SS + c0]; b0.y = W[(kb + 1) * SS + c0];
      acc[g][0] = wmma_f32(a, b0, acc[g][0]);
      v2f b1; b1.x = W[kb * SS + c1]; b1.y = W[(kb + 1) * SS + c1];
      acc[g][1] = wmma_f32(a, b1, acc[g][1]);
    }
  }
  for (int kk = 0; kk < SS; kk += 4) {  // H0 @ Th_g
    const int kb = kk + krow;
    v2f a; a.x = Gs[arow][kb]; a.y = Gs[arow][kb + 1];
#pragma unroll
    for (int g = 0; g < 4; g++) {
      const float* T = Tg[g];
      v2f b0; b0.x = T[kb * SS + c0]; b0.y = T[(kb + 1) * SS + c0];
      acc[g][0] = wmma_f32(a, b0, acc[g][0]);
      v2f b1; b1.x = T[kb * SS + c1]; b1.y = T[(kb + 1) * SS + c1];
      acc[g][1] = wmma_f32(a, b1, acc[g][1]);
    }
  }

  const int rbase = 8 * (lane >> 4);
#pragma unroll
  for (int t = 0; t < 2; t++) {
    const int c = (t == 0) ? c0 : c1;
    const float bias_i = bci[c] + bi[c];
    const float bias_f = bcf[c] + bfg[c];
    const float bias_c = bcc[c] + bcg[c];
    const float bias_o = bco[c] + bo[c];
#pragma unroll
    for (int r = 0; r < 8; r++) {
      const int row = row0 + rbase + r;
      const float iv = sigf(acc[0][t][r] + bias_i);
      const float fv = sigf(acc[1][t][r] + bias_f);
      const float tv = tanhf(acc[2][t][r] + bias_c);
      const float ov = sigf(acc[3][t][r] + bias_o);
      const float c0v = C0[(size_t)row * SS + c];
      const float cn = fv * c0v + iv * tv;
      const float hn = ov * tanhf(cn);
      CnOut[(size_t)row * SS + c] = cn;
      HnOut[(size_t)row * SS + c] = hn;
    }
  }
}

// ---------------- logits = relu(Hn) @ Wl + bl ----------------
__global__ __launch_bounds__(256) void k_logits(const float* __restrict__ Hn,
                                                const float* __restrict__ Wl,
                                                const float* __restrict__ bl,
                                                float* __restrict__ out) {
  const int lane = threadIdx.x & 31, wv = threadIdx.x >> 5;
  const int rt = blockIdx.x * 8 + wv;
  if (rt >= NN / 16) return;  // wave-uniform branch, EXEC stays full
  const int row0 = rt * 16;
  const int arow = lane & 15;
  const int krow = 2 * (lane >> 4);
  const int c0 = lane & 15, c1 = c0 + 16;
  v8f a0 = {}, a1 = {};
  for (int kk = 0; kk < SS; kk += 4) {
    const int kb = kk + krow;
    const float* hp = Hn + (size_t)(row0 + arow) * SS + kb;
    v2f a; a.x = fmaxf(hp[0], 0.0f); a.y = fmaxf(hp[1], 0.0f);
    v2f b0; b0.x = Wl[kb * CC + c0]; b0.y = Wl[(kb + 1) * CC + c0];
    a0 = wmma_f32(a, b0, a0);
    v2f b1; b1.x = Wl[kb * CC + c1]; b1.y = Wl[(kb + 1) * CC + c1];
    a1 = wmma_f32(a, b1, a1);
  }
  const int rbase = 8 * (lane >> 4);
  const float bl0 = bl[c0], bl1 = bl[c1];
#pragma unroll
  for (int r = 0; r < 8; r++) {
    const int row = row0 + rbase + r;
    out[(size_t)row * CC + c0] = a0[r] + bl0;
    out[(size_t)row * CC + c1] = a1[r] + bl1;
  }
}

// ---------------- row softmax over 32 classes ----------------
__global__ __launch_bounds__(256) void k_softmax(float* out) {
  const int row = blockIdx.x * 8 + (threadIdx.x >> 5);
  const int lane = threadIdx.x & 31;
  float v = out[(size_t)row * CC + lane];
  float m = v;
  for (int off = 16; off >= 1; off >>= 1) m = fmaxf(m, __shfl_xor(m, off, 32));
  float e = __expf(v - m);
  float s = e;
  for (int off = 16; off >= 1; off >>= 1) s += __shfl_xor(s, off, 32);
  out[(size_t)row * CC + lane] = e / s;
}

extern "C" void kernel_launch(void* const* d_in, const int* in_sizes, int n_in,
                              void* d_out, int out_size, void* d_ws,
                              size_t ws_size, hipStream_t stream) {
  const float* x = (const float*)d_in[0];
  const int* ei = (const int*)d_in[1];
  const float* ew = (const float*)d_in[2];
  const float* H0 = (const float*)d_in[3];
  const float* C0 = (const float*)d_in[4];
  const float* W1 = (const float*)d_in[5];
  const float* b1 = (const float*)d_in[6];
  const float* W_i = (const float*)d_in[7];
  const float* Th_i = (const float*)d_in[8];
  const float* bc_i = (const float*)d_in[9];
  const float* b_i = (const float*)d_in[10];
  const float* W_f = (const float*)d_in[11];
  const float* Th_f = (const float*)d_in[12];
  const float* bc_f = (const float*)d_in[13];
  const float* b_f = (const float*)d_in[14];
  const float* W_c = (const float*)d_in[15];
  const float* Th_c = (const float*)d_in[16];
  const float* bc_c = (const float*)d_in[17];
  const float* b_c = (const float*)d_in[18];
  const float* W_o = (const float*)d_in[19];
  const float* Th_o = (const float*)d_in[20];
  const float* bc_o = (const float*)d_in[21];
  const float* b_o = (const float*)d_in[22];
  const float* Wl = (const float*)d_in[23];
  const float* bl = (const float*)d_in[24];

  const int* src = ei;
  const int* dst = ei + EE;

  // d_out layout (return order): [softmax N*C][Hn N*S][Cn N*S]
  float* outSm = (float*)d_out;
  float* HnOut = outSm + (size_t)NN * CC;
  float* CnOut = HnOut + (size_t)NN * SS;

  // workspace: deg/dinv (N) + h (N*F).  xw is staged in the Cn region of
  // d_out (N*F <= N*S) and fully overwritten by k_gates afterwards.
  float* deg = (float*)d_ws;
  float* h = deg + NN;
  float* xw = CnOut;

  k_deg_init<<<(NN + 255) / 256, 256, 0, stream>>>(deg);
  k_deg_acc<<<(EE + 255) / 256, 256, 0, stream>>>(dst, ew, deg);
  k_dinv<<<(NN + 255) / 256, 256, 0, stream>>>(deg);
  k_gemm_xw<<<NN / 16, 256, 0, stream>>>(x, W1, xw);
  k_h_init<<<(NN * FF + 255) / 256, 256, 0, stream>>>(xw, deg, b1, h);
  k_scatter<<<EE / 8, 256, 0, stream>>>(src, dst, ew, deg, xw, h);
  k_gates<<<NN / 16, 256, 0, stream>>>(h, H0, C0, W_i, W_f, W_c, W_o, Th_i,
                                       Th_f, Th_c, Th_o, bc_i, bc_f, bc_c,
                                       bc_o, b_i, b_f, b_c, b_o, HnOut, CnOut);
  k_logits<<<(NN / 16 + 7) / 8, 256, 0, stream>>>(HnOut, Wl, bl, outSm);
  k_softmax<<<NN / 8, 256, 0, stream>>>(outSm);
}